// ROIPooler_13005160972850
// MI455X (gfx1250) — compile-verified
//
#include <hip/hip_runtime.h>

typedef __attribute__((ext_vector_type(2))) float v2f;
typedef __attribute__((ext_vector_type(8))) float v8f;

#define Nn 2
#define Cc 256
#define Hh 200
#define Ww 200
#define Mm 512
#define Pp 14

// ---------------------------------------------------------------------------
// Kernel 1: NCHW -> NHWC transpose via LDS tile (coalesced read and write).
// x[(n*C + c)*H*W + h*W + w]  ->  y[((n*H + h)*W + w)*C + c]
// ---------------------------------------------------------------------------
__global__ __launch_bounds__(256) void nchw_to_nhwc(const float* __restrict__ x,
                                                    float* __restrict__ y) {
    __shared__ float tile[32][33];
    const int w0 = blockIdx.x * 32;
    const int c0 = blockIdx.y * 32;
    const int nh = blockIdx.z;            // n*H + h
    const int n  = nh / Hh;
    const int h  = nh % Hh;
    const int tx = threadIdx.x;           // 0..31
    const int ty = threadIdx.y;           // 0..7

#pragma unroll
    for (int i = 0; i < 4; ++i) {
        int c = c0 + ty + 8 * i;
        int w = w0 + tx;
        if (c < Cc && w < Ww)
            tile[ty + 8 * i][tx] =
                x[(((long)n * Cc + c) * Hh + h) * (long)Ww + w];
    }
    __syncthreads();
#pragma unroll
    for (int i = 0; i < 4; ++i) {
        int w = w0 + ty + 8 * i;
        int c = c0 + tx;
        if (w < Ww && c < Cc)
            y[(((long)n * Hh + h) * (long)Ww + w) * Cc + c] = tile[tx][ty + 8 * i];
    }
}

// ---------------------------------------------------------------------------
// Kernel 2: ROI-Align as separable per-box matmuls using V_WMMA_F32_16X16X4_F32.
// Grid: (Pp, Mm) = (bin-row ph, box m). Block: 512 threads = 16 waves,
// wave w owns channel tile [16w, 16w+16).
//
// out[c,ph,pw] = sum_{yc in 4 y-corners} sum_{j in 56 x-corners}
//                   Wy[yc] * Wx[j] * X[n, yrow[yc], xcol[j], c]
// Inner x-contraction: D(16c x 16pw) += A(16c x 4) * B(4 x 16pw), 14 rank-4
// k-steps (j = 4*j0+k has pw == j0, so B is nonzero only in column j0).
// Weights carry validity masks and the 1/4 sub-sample mean; invalid -> 0.
//
// Strides are template constants (all offsets fit in 32 bits) and the row
// base is scalarized with readfirstlane so loads use SGPR-base + 32-bit
// VGPR offset (GVS) addressing instead of per-lane 64-bit address math.
// ---------------------------------------------------------------------------
template <int SN, int SY, int SX, int SC>
__global__ __launch_bounds__(512) void roi_align_wmma(const float* __restrict__ src,
                                                      const float* __restrict__ boxes,
                                                      float* __restrict__ out) {
    __shared__ float s_wx[56];
    __shared__ int   s_xc[56];
    __shared__ float s_wy[4];
    __shared__ int   s_yr[4];
    __shared__ int   s_n;

    const int ph  = blockIdx.x;
    const int m   = blockIdx.y;
    const int tid = threadIdx.x;

    if (tid < 28) {                       // x-axis sample s = tid -> corners 2s, 2s+1
        const int s = tid;
        float bx0 = boxes[m * 5 + 1] * 0.25f;
        float bx1 = boxes[m * 5 + 3] * 0.25f;
        float binw = fmaxf(bx1 - bx0, 1.0f) * (1.0f / Pp);
        float off  = (float)(s >> 1) + ((float)(s & 1) + 0.5f) * 0.5f;
        float pos  = bx0 + off * binw;
        bool  valid = (pos > -1.0f) && (pos < (float)Ww);
        float cpos  = fminf(fmaxf(pos, 0.0f), (float)(Ww - 1));
        int   lo    = (int)floorf(cpos);
        int   hi    = min(lo + 1, Ww - 1);
        float fr    = cpos - (float)lo;
        s_xc[2 * s]     = lo;
        s_xc[2 * s + 1] = hi;
        s_wx[2 * s]     = valid ? 0.5f * (1.0f - fr) : 0.0f;  // 0.5 = x half of mean
        s_wx[2 * s + 1] = valid ? 0.5f * fr          : 0.0f;
    } else if (tid < 30) {                // y-axis samples for this bin row
        const int q = tid - 28;           // sub-sample 0/1
        const int s = 2 * ph + q;
        float by0 = boxes[m * 5 + 2] * 0.25f;
        float by1 = boxes[m * 5 + 4] * 0.25f;
        float binh = fmaxf(by1 - by0, 1.0f) * (1.0f / Pp);
        float off  = (float)(s >> 1) + ((float)(s & 1) + 0.5f) * 0.5f;
        float pos  = by0 + off * binh;
        bool  valid = (pos > -1.0f) && (pos < (float)Hh);
        float cpos  = fminf(fmaxf(pos, 0.0f), (float)(Hh - 1));
        int   lo    = (int)floorf(cpos);
        int   hi    = min(lo + 1, Hh - 1);
        float fr    = cpos - (float)lo;
        s_yr[2 * q]     = lo;
        s_yr[2 * q + 1] = hi;
        s_wy[2 * q]     = valid ? 0.5f * (1.0f - fr) : 0.0f;  // 0.5 = y half of mean
        s_wy[2 * q + 1] = valid ? 0.5f * fr          : 0.0f;
    } else if (tid == 30) {
        s_n = (int)boxes[m * 5 + 0];
    }
    __syncthreads();

    const int lane = tid & 31;
    const int colm = lane & 15;           // A: M (channel-in-tile); B/D: N (pw)
    const int kh   = lane >> 4;           // K-half select (K=0,1 vs K=2,3)
    // Wave-uniform values -> SGPRs so load addressing is SGPR-base + 32b offset.
    const int c0   = __builtin_amdgcn_readfirstlane((tid >> 5) * 16);
    const int n    = __builtin_amdgcn_readfirstlane(s_n);

    const float* bs = src + (long)n * SN + (long)c0 * SC;   // scalar base
    const int coff  = colm * SC;                            // per-lane channel offset

    v8f acc0 = {0.0f, 0.0f, 0.0f, 0.0f, 0.0f, 0.0f, 0.0f, 0.0f};
    v8f acc1 = {0.0f, 0.0f, 0.0f, 0.0f, 0.0f, 0.0f, 0.0f, 0.0f};

#pragma unroll
    for (int yc = 0; yc < 4; ++yc) {
        const int   yr = __builtin_amdgcn_readfirstlane(s_yr[yc]);
        const float wy = s_wy[yc];
        const float* rb = bs + (long)yr * SY;               // scalar row base
        if (yc < 3) {   // pull the next row toward L2/L0 while contracting this one
            const int ynext = __builtin_amdgcn_readfirstlane(s_yr[yc + 1]);
            __builtin_prefetch(bs + (long)ynext * SY + lane, 0, 1);
        }
#pragma unroll
        for (int j0 = 0; j0 < 14; ++j0) {
            const int ka = 4 * j0 + 2 * kh;
            v2f a, b;
            // A fragment: 16x4 f32; lanes 0-15 hold K=0,1, lanes 16-31 hold K=2,3.
            a.x = rb[(unsigned)(s_xc[ka]     * SX + coff)];
            a.y = rb[(unsigned)(s_xc[ka + 1] * SX + coff)];
            // B fragment: 4x16, nonzero only in column pw == j0.
            const float g = (colm == j0) ? wy : 0.0f;
            b.x = g * s_wx[ka];
            b.y = g * s_wx[ka + 1];
            if (yc & 2)
                acc1 = __builtin_amdgcn_wmma_f32_16x16x4_f32(
                    false, a, false, b, (short)0, acc1, false, false);
            else
                acc0 = __builtin_amdgcn_wmma_f32_16x16x4_f32(
                    false, a, false, b, (short)0, acc0, false, false);
        }
    }

    const v8f acc = acc0 + acc1;

    // D layout: VGPR r, lanes 0-15 -> M=r, N=lane; lanes 16-31 -> M=r+8, N=lane-16.
    if (colm < Pp) {
        float* ob = out + (((long)m * Cc + c0) * Pp + ph) * Pp;  // scalar base
#pragma unroll
        for (int r = 0; r < 8; ++r) {
            const int Mrow = r + 8 * kh;   // channel within tile
            ob[(unsigned)(Mrow * (Pp * Pp) + colm)] = acc[r];
        }
    }
}

// ---------------------------------------------------------------------------
extern "C" void kernel_launch(void* const* d_in, const int* in_sizes, int n_in,
                              void* d_out, int out_size, void* d_ws, size_t ws_size,
                              hipStream_t stream) {
    const float* x     = (const float*)d_in[0];   // (2,256,200,200) f32
    const float* boxes = (const float*)d_in[1];   // (512,5) f32
    float*       out   = (float*)d_out;           // (512,256,14,14) f32

    const size_t x_bytes = (size_t)Nn * Cc * Hh * Ww * sizeof(float); // 81.92 MB

    if (ws_size >= x_bytes) {
        float* nhwc = (float*)d_ws;
        nchw_to_nhwc<<<dim3((Ww + 31) / 32, Cc / 32, Nn * Hh), dim3(32, 8), 0, stream>>>(x, nhwc);
        // NHWC strides: sn = H*W*C, sy = W*C, sx = C, sc = 1 (channel-coalesced gathers)
        roi_align_wmma<Hh * Ww * Cc, Ww * Cc, Cc, 1>
            <<<dim3(Pp, Mm), 512, 0, stream>>>(nhwc, boxes, out);
    } else {
        // Fallback: gather straight from NCHW (uncoalesced but correct).
        roi_align_wmma<Cc * Hh * Ww, Ww, 1, Hh * Ww>
            <<<dim3(Pp, Mm), 512, 0, stream>>>(x, boxes, out);
    }
}